// RNNCFModel_84421877170767
// MI455X (gfx1250) — compile-verified
//
#include <hip/hip_runtime.h>
#include <stddef.h>

// ---------------------------------------------------------------------------
// LSTM car-following rollout for MI455X (gfx1250, wave32).
// One wave handles a 16-vehicle WMMA tile for all 1024 timesteps.
// Gate GEMM z = [inp|h] @ [kernel;rkernel] done with V_WMMA_F32_16X16X4_F32
// (fp32 in/out -> matches reference numerics).  45 WMMAs / wave / step.
// Per-wave private LDS slices -> no barriers inside the time loop.
// ---------------------------------------------------------------------------

typedef float v2f __attribute__((ext_vector_type(2)));
typedef float v8f __attribute__((ext_vector_type(8)));

#define NVEH  4096
#define NT    1024
#define INDIM 12
#define UNITS 20

__device__ __forceinline__ v8f wmma4(v2f a, v2f b, v8f c) {
  // 8-arg pattern: (neg_a, A, neg_b, B, c_mod, C, reuse_a, reuse_b)
  return __builtin_amdgcn_wmma_f32_16x16x4_f32(false, a, false, b, (short)0, c,
                                               false, false);
}

__device__ __forceinline__ float sigf(float x) {
  return __fdividef(1.0f, 1.0f + __expf(-x));
}
__device__ __forceinline__ float tanhf_fast(float x) {
  return 1.0f - __fdividef(2.0f, 1.0f + __expf(2.0f * x));
}
__device__ __forceinline__ float nanfix(float v) { return (v != v) ? 1.0f : v; }

__global__ __launch_bounds__(256) void rnncf_kernel(
    const float* __restrict__ leadfol,   // (NVEH, NT, 12)
    const float* __restrict__ init_state,// (NVEH, 2)
    const float* __restrict__ hidden,    // (2, NVEH, 20)
    const float* __restrict__ Wk,        // (12, 80)
    const float* __restrict__ Wr,        // (20, 80)
    const float* __restrict__ bias,      // (80)
    const float* __restrict__ W2,        // (20, 10)
    const float* __restrict__ b2,        // (10)
    const float* __restrict__ Wlc,       // (10, 3)
    const float* __restrict__ blc,       // (3)
    const float* __restrict__ W1,        // (10, 1)
    const float* __restrict__ b1,        // (1)
    float* __restrict__ traj,            // (NVEH, NT)
    float* __restrict__ lcout,           // (NVEH, NT, 3)
    float* __restrict__ spdout,          // (NVEH)
    float* __restrict__ hout,            // (NVEH, 20)
    float* __restrict__ cout)            // (NVEH, 20)
{
  // 44 shared floats (lc/acc heads) + 8 per-wave slices of 1952 floats.
  __shared__ float smem[44 + 8 * 1952];  // 62,640 B

  const int tid  = threadIdx.x;
  const int wid  = tid >> 5;
  const int lane = tid & 31;
  const int n    = lane & 15;  // WMMA column / vehicle row (A-side)
  const int half = lane >> 4;  // 0: lanes 0-15, 1: lanes 16-31
  const int m    = n;          // vehicle index within the 16-wide tile
  const int vb   = blockIdx.x * 128 + wid * 16;  // first vehicle of this wave

  float* wlc_s = smem;        // 30
  float* blc_s = smem + 30;   // 3
  float* w1_s  = smem + 33;   // 10
  float* b1_s  = smem + 43;   // 1
  float* wbase = smem + 44 + wid * 1952;
  float* zbuf  = wbase;          // 16 x 84 (padded stride vs 80 cols)
  float* hbuf  = wbase + 1344;   // 16 x 20
  float* x2b   = hbuf + 320;     // 16 x 17 (padded stride vs 16 cols)
  float* posb  = x2b + 272;      // 16

  if (tid < 30)       wlc_s[tid]      = Wlc[tid];
  else if (tid < 33)  blc_s[tid - 30] = blc[tid - 30];
  else if (tid < 43)  w1_s[tid - 33]  = W1[tid - 33];
  else if (tid == 43) b1_s[0]         = b1[0];

  // ---- register-resident B fragments (loaded once) ----------------------
  // B 4x16 layout: VGPR0 = row K0 (lanes 0-15) / K0+2 (lanes 16-31), VGPR1 = +1
  v2f kB[5][3], rB[5][5], w2B[5];
  float biasr[5];
#pragma unroll
  for (int j = 0; j < 5; ++j) {
    biasr[j] = bias[j * 16 + n];
#pragma unroll
    for (int kc = 0; kc < 3; ++kc) {
      const int k0 = kc * 4 + 2 * half;
      v2f b; b.x = Wk[k0 * 80 + j * 16 + n]; b.y = Wk[(k0 + 1) * 80 + j * 16 + n];
      kB[j][kc] = b;
    }
#pragma unroll
    for (int kc = 0; kc < 5; ++kc) {
      const int k0 = kc * 4 + 2 * half;
      v2f b; b.x = Wr[k0 * 80 + j * 16 + n]; b.y = Wr[(k0 + 1) * 80 + j * 16 + n];
      rB[j][kc] = b;
    }
  }
  const float b2r = (n < 10) ? b2[n] : 0.0f;
#pragma unroll
  for (int kc = 0; kc < 5; ++kc) {
    const int k0 = kc * 4 + 2 * half;
    v2f b;
    b.x = (n < 10) ? W2[k0 * 10 + n] : 0.0f;
    b.y = (n < 10) ? W2[(k0 + 1) * 10 + n] : 0.0f;
    w2B[kc] = b;
  }

  // ---- state init -------------------------------------------------------
  float pos = 0.0f, spd = 0.0f;
  if (half == 0) {
    pos = init_state[(vb + m) * 2 + 0];
    spd = init_state[(vb + m) * 2 + 1];
    posb[m] = pos;
  }

  float cr[10];
  int   zoff[10], hq[10];
#pragma unroll
  for (int p = 0; p < 10; ++p) {
    const int q  = p * 32 + lane;      // 0..319  == veh*20 + unit
    const int vr = q / 20, un = q % 20;
    hq[p]   = q;
    zoff[p] = vr * 84 + un;
    hbuf[q] = hidden[vb * UNITS + q];                 // h0
    cr[p]   = hidden[NVEH * UNITS + vb * UNITS + q];  // c0
  }
  __syncthreads();  // only for the shared lc/acc-head weights

  // A fragments of h (16x4 f32 layout: lanes 0-15 K0..K0+1, lanes 16-31 +2)
  v2f hA[5];
#pragma unroll
  for (int kc = 0; kc < 5; ++kc) {
    const float2 hv =
        *reinterpret_cast<const float2*>(&hbuf[m * 20 + kc * 4 + 2 * half]);
    v2f a; a.x = hv.x; a.y = hv.y; hA[kc] = a;
  }

  const float invHD = 1.0f / 100.0f, invV = 1.0f / 40.0f;
  const float* xptr = leadfol + (size_t)(vb + m) * NT * INDIM + 2 * half;

  for (int t = 0; t < NT; ++t) {
    // ---- feature build: lane covers k = {2h,2h+1, 4+2h,4+2h+1, 8+2h,8+2h+1}
    const float2 xa = *reinterpret_cast<const float2*>(xptr);
    const float2 xb = *reinterpret_cast<const float2*>(xptr + 4);
    const float2 xc = *reinterpret_cast<const float2*>(xptr + 8);
    if (t + 1 < NT) __builtin_prefetch(xptr + INDIM, 0, 0);
    const float posm = posb[m];

    v2f iA[3];
    iA[0].x = nanfix((xa.x - posm) * invHD);                              // k=0|2 lead
    iA[0].y = nanfix(half ? (posm - xa.y) * invHD : (xa.y - posm) * invHD); // k=1 lead | k=3 fol
    iA[1].x = nanfix(half ? xb.x * invV : (posm - xb.x) * invHD);         // k=4 fol | k=6 spd
    iA[1].y = nanfix(half ? xb.y * invV : (posm - xb.y) * invHD);         // k=5 fol | k=7 spd
    iA[2].x = nanfix(xc.x * invV);                                        // k=8|10 spd
    iA[2].y = nanfix(xc.y * invV);                                        // k=9|11 spd

    // ---- z = inp@Wk + h@Wr + bias  (5 N-tiles x 8 K-chunks of WMMA) ----
#pragma unroll
    for (int j = 0; j < 5; ++j) {
      v8f acc;
#pragma unroll
      for (int r = 0; r < 8; ++r) acc[r] = biasr[j];
#pragma unroll
      for (int kc = 0; kc < 3; ++kc) acc = wmma4(iA[kc], kB[j][kc], acc);
#pragma unroll
      for (int kc = 0; kc < 5; ++kc) acc = wmma4(hA[kc], rB[j][kc], acc);
#pragma unroll
      for (int r = 0; r < 8; ++r)
        zbuf[(r + 8 * half) * 84 + j * 16 + n] = acc[r];  // D: row r+8*half
    }

    // ---- LSTM elementwise (10 (veh,unit) pairs per lane) ---------------
#pragma unroll
    for (int p = 0; p < 10; ++p) {
      const int zo = zoff[p];
      const float ig = zbuf[zo +  0];
      const float fg = zbuf[zo + 20];
      const float gg = zbuf[zo + 40];
      const float og = zbuf[zo + 60];
      const float cn = sigf(fg) * cr[p] + sigf(ig) * tanhf_fast(gg);
      cr[p] = cn;
      hbuf[hq[p]] = sigf(og) * tanhf_fast(cn);
    }

    // ---- rebuild h A-fragments (also serve next iteration) -------------
#pragma unroll
    for (int kc = 0; kc < 5; ++kc) {
      const float2 hv =
          *reinterpret_cast<const float2*>(&hbuf[m * 20 + kc * 4 + 2 * half]);
      v2f a; a.x = hv.x; a.y = hv.y; hA[kc] = a;
    }

    // ---- x2 = relu(h@W2 + b2)  (one padded N-tile, 5 WMMAs) ------------
    {
      v8f acc;
#pragma unroll
      for (int r = 0; r < 8; ++r) acc[r] = b2r;
#pragma unroll
      for (int kc = 0; kc < 5; ++kc) acc = wmma4(hA[kc], w2B[kc], acc);
#pragma unroll
      for (int r = 0; r < 8; ++r) {
        const float v = acc[r];
        x2b[(r + 8 * half) * 17 + n] = v > 0.0f ? v : 0.0f;
      }
    }

    // ---- lc / acc heads + vehicle dynamics (lanes 0-15, 1 veh each) ----
    if (half == 0) {
      float lc0 = blc_s[0], lc1 = blc_s[1], lc2 = blc_s[2], ar = b1_s[0];
#pragma unroll
      for (int u = 0; u < 10; ++u) {
        const float xv = x2b[m * 17 + u];
        lc0 += xv * wlc_s[u * 3 + 0];
        lc1 += xv * wlc_s[u * 3 + 1];
        lc2 += xv * wlc_s[u * 3 + 2];
        ar  += xv * w1_s[u];
      }
      const float accv = 10.0f * ar - 6.0f;  // (MAXA-MINA)*ar + MINA
      pos = pos + 0.1f * spd;
      spd = spd + 0.1f * accv;
      posb[m] = pos;
      const int vg = vb + m;
      traj[(size_t)vg * NT + t] = pos;
      float* lp = lcout + ((size_t)vg * NT + t) * 3;
      lp[0] = lc0; lp[1] = lc1; lp[2] = lc2;
    }
    xptr += INDIM;
  }

  // ---- finals -----------------------------------------------------------
  if (half == 0) spdout[vb + m] = spd;
#pragma unroll
  for (int p = 0; p < 10; ++p) {
    hout[vb * UNITS + hq[p]] = hbuf[hq[p]];
    cout[vb * UNITS + hq[p]] = cr[p];
  }
}

extern "C" void kernel_launch(void* const* d_in, const int* in_sizes, int n_in,
                              void* d_out, int out_size, void* d_ws,
                              size_t ws_size, hipStream_t stream) {
  (void)in_sizes; (void)n_in; (void)out_size; (void)d_ws; (void)ws_size;
  const float* leadfol    = (const float*)d_in[0];
  const float* init_state = (const float*)d_in[1];
  const float* hidden     = (const float*)d_in[2];
  const float* Wk         = (const float*)d_in[3];
  const float* Wr         = (const float*)d_in[4];
  const float* bias       = (const float*)d_in[5];
  const float* W2         = (const float*)d_in[6];
  const float* b2         = (const float*)d_in[7];
  const float* Wlc        = (const float*)d_in[8];
  const float* blc        = (const float*)d_in[9];
  const float* W1         = (const float*)d_in[10];
  const float* b1         = (const float*)d_in[11];

  float* out  = (float*)d_out;
  float* traj = out;                                   // 4096*1024
  float* lc   = out + (size_t)NVEH * NT;               // 4096*1024*3
  float* spdf = lc  + (size_t)NVEH * NT * 3;           // 4096
  float* hf   = spdf + NVEH;                           // 4096*20
  float* cf   = hf   + (size_t)NVEH * UNITS;           // 4096*20

  rnncf_kernel<<<NVEH / 128, 256, 0, stream>>>(
      leadfol, init_state, hidden, Wk, Wr, bias, W2, b2, Wlc, blc, W1, b1,
      traj, lc, spdf, hf, cf);
}